// GCNEncoder_36644660970265
// MI455X (gfx1250) — compile-verified
//
#include <hip/hip_runtime.h>

#define DIM_IN 256
#define DIM_H  128
#define DIM_Z  32
#define LN_EPS 1e-5f

typedef __attribute__((ext_vector_type(2))) float v2f;
typedef __attribute__((ext_vector_type(8))) float v8f;

// ---------------- degree / normalization ----------------

__global__ void degree_kernel(const int* __restrict__ dst, float* __restrict__ deg, int E) {
    int i = blockIdx.x * blockDim.x + threadIdx.x;
    if (i < E) atomicAdd(&deg[dst[i]], 1.0f);
}

// deg -> dis = rsqrt(deg + 1)  (in place)
__global__ void rsqrt_kernel(float* __restrict__ deg, int n) {
    int i = blockIdx.x * blockDim.x + threadIdx.x;
    if (i < n) deg[i] = rsqrtf(deg[i] + 1.0f);
}

// ---------------- layer norm: one wave32 per 256-wide row ----------------

__global__ void layernorm_kernel(const float* __restrict__ x,
                                 const float* __restrict__ w,
                                 const float* __restrict__ b,
                                 float* __restrict__ out, int n) {
    int gtid = blockIdx.x * blockDim.x + threadIdx.x;
    int row  = gtid >> 5;           // wave32: 32 lanes per row
    int lane = threadIdx.x & 31;
    if (row >= n) return;

    const float4* xr = (const float4*)(x + (size_t)row * DIM_IN);
    float4 v0 = xr[lane];           // elements [lane*4 .. lane*4+3]
    float4 v1 = xr[lane + 32];      // elements [128 + lane*4 ..]

    float s = v0.x + v0.y + v0.z + v0.w + v1.x + v1.y + v1.z + v1.w;
#pragma unroll
    for (int off = 16; off > 0; off >>= 1) s += __shfl_xor(s, off, 32);
    float mu = s * (1.0f / DIM_IN);

    float sv = 0.f, d;
    d = v0.x - mu; sv += d * d;  d = v0.y - mu; sv += d * d;
    d = v0.z - mu; sv += d * d;  d = v0.w - mu; sv += d * d;
    d = v1.x - mu; sv += d * d;  d = v1.y - mu; sv += d * d;
    d = v1.z - mu; sv += d * d;  d = v1.w - mu; sv += d * d;
#pragma unroll
    for (int off = 16; off > 0; off >>= 1) sv += __shfl_xor(sv, off, 32);
    float inv = rsqrtf(sv * (1.0f / DIM_IN) + LN_EPS);

    const float4* w4 = (const float4*)w;
    const float4* b4 = (const float4*)b;
    float4 W0 = w4[lane], Wb = w4[lane + 32];
    float4 B0 = b4[lane], Bb = b4[lane + 32];
    float4 o0, o1;
    o0.x = (v0.x - mu) * inv * W0.x + B0.x;
    o0.y = (v0.y - mu) * inv * W0.y + B0.y;
    o0.z = (v0.z - mu) * inv * W0.z + B0.z;
    o0.w = (v0.w - mu) * inv * W0.w + B0.w;
    o1.x = (v1.x - mu) * inv * Wb.x + Bb.x;
    o1.y = (v1.y - mu) * inv * Wb.y + Bb.y;
    o1.z = (v1.z - mu) * inv * Wb.z + Bb.z;
    o1.w = (v1.w - mu) * inv * Wb.w + Bb.w;
    float4* orow = (float4*)(out + (size_t)row * DIM_IN);
    orow[lane]      = o0;
    orow[lane + 32] = o1;
}

// ---------------- fp32 WMMA GEMM: C[M,Nout] = A[M,K] @ B[K,Nout] ----------------
// block = 128 threads (4 waves). wave w -> 16x16 tile at (blockIdx.x*16, (blockIdx.y*4+w)*16)
// V_WMMA_F32_16X16X4_F32 layouts (ISA 7.12.2):
//   A 16x4: lanes 0-15 hold M=lane, v0=K+0,v1=K+1; lanes 16-31 hold v0=K+2,v1=K+3
//   B 4x16 / C,D 16x16: rows striped across lanes within a VGPR

__global__ void wmma_gemm_f32(const float* __restrict__ A, const float* __restrict__ B,
                              float* __restrict__ C, int K, int Nout) {
    const int waveId = threadIdx.x >> 5;
    const int lane   = threadIdx.x & 31;
    const int half   = lane >> 4;    // 0: lanes 0-15, 1: lanes 16-31
    const int l      = lane & 15;

    const int rowBase = blockIdx.x * 16;
    const int colBase = (blockIdx.y * 4 + waveId) * 16;
    if (colBase >= Nout) return;     // wave-uniform: EXEC stays all-1s for WMMA

    const float* Arow = A + (size_t)(rowBase + l) * K;
    v8f c = {};
#pragma unroll 4
    for (int k = 0; k < K; k += 4) {
        const int ka = k + 2 * half;           // even -> 8B aligned
        v2f a = *(const v2f*)(Arow + ka);      // A[row][ka], A[row][ka+1]
        v2f b;
        b.x = B[(size_t)ka * Nout + colBase + l];
        b.y = B[(size_t)(ka + 1) * Nout + colBase + l];
        c = __builtin_amdgcn_wmma_f32_16x16x4_f32(false, a, false, b,
                                                  (short)0, c, false, false);
    }
    // C VGPR j: lanes 0-15 -> row j, lanes 16-31 -> row 8+j
    float* Crow = C + (size_t)(rowBase + half * 8) * Nout + colBase + l;
#pragma unroll
    for (int j = 0; j < 8; ++j) Crow[(size_t)j * Nout] = c[j];
}

// ---------------- aggregation ----------------

// out[i,f] = t[i,f] * dis[i]^2 + bias[f]   (self-loop term + bias, also inits the atomic target)
__global__ void selfloop_bias_kernel(const float* __restrict__ t, const float* __restrict__ dis,
                                     const float* __restrict__ bias, float* __restrict__ out,
                                     int F, size_t total) {
    size_t idx = (size_t)blockIdx.x * blockDim.x + threadIdx.x;
    if (idx >= total) return;
    size_t i = idx / F;
    int    f = (int)(idx - i * F);
    float  di = dis[i];
    out[idx] = t[idx] * di * di + bias[f];
}

// one wave32 per edge; lanes stride the feature dim; atomic scatter-add to dst row
__global__ void edge_scatter_kernel(const float* __restrict__ t,
                                    const int* __restrict__ src, const int* __restrict__ dst,
                                    const float* __restrict__ dis, float* __restrict__ out,
                                    int E, int F) {
    int e    = (int)(((size_t)blockIdx.x * blockDim.x + threadIdx.x) >> 5);
    int lane = threadIdx.x & 31;
    if (e >= E) return;
    int s = src[e], d = dst[e];
    float nrm = dis[s] * dis[d];
    const float* trow = t + (size_t)s * F;
    float* orow = out + (size_t)d * F;
    for (int f = lane; f < F; f += 32)
        atomicAdd(&orow[f], trow[f] * nrm);
}

__global__ void relu_kernel(float* __restrict__ a, size_t total) {
    size_t idx = (size_t)blockIdx.x * blockDim.x + threadIdx.x;
    if (idx < total) a[idx] = fmaxf(a[idx], 0.0f);
}

// ---------------- launch ----------------

extern "C" void kernel_launch(void* const* d_in, const int* in_sizes, int n_in,
                              void* d_out, int out_size, void* d_ws, size_t ws_size,
                              hipStream_t stream) {
    const float* x    = (const float*)d_in[0];
    const int*   ei   = (const int*)d_in[1];
    const float* ln_w = (const float*)d_in[2];
    const float* ln_b = (const float*)d_in[3];
    const float* W1   = (const float*)d_in[4];
    const float* b1   = (const float*)d_in[5];
    const float* W2   = (const float*)d_in[6];
    const float* b2   = (const float*)d_in[7];
    const float* W3   = (const float*)d_in[8];
    const float* b3   = (const float*)d_in[9];

    const int n = in_sizes[0] / DIM_IN;   // 65536
    const int E = in_sizes[1] / 2;        // 524288
    const int* src = ei;                  // edge_index[0]
    const int* dst = ei + E;              // edge_index[1]

    float* dis = (float*)d_ws;                    // n
    float* h0  = dis + n;                         // n * 256
    float* t   = h0 + (size_t)n * DIM_IN;         // n * 128 (reused per layer)
    float* g   = t  + (size_t)n * DIM_H;          // n * 128 (reused per layer)
    float* out = (float*)d_out;                   // n * 32

    // degrees -> dis = rsqrt(deg + 1)
    hipMemsetAsync(dis, 0, (size_t)n * sizeof(float), stream);
    degree_kernel<<<(E + 255) / 256, 256, 0, stream>>>(dst, dis, E);
    rsqrt_kernel<<<(n + 255) / 256, 256, 0, stream>>>(dis, n);

    // layer norm
    layernorm_kernel<<<(n * 32 + 255) / 256, 256, 0, stream>>>(x, ln_w, ln_b, h0, n);

    const size_t totH = (size_t)n * DIM_H;
    const size_t totZ = (size_t)n * DIM_Z;
    dim3 gemmGridH(n / 16, DIM_H / 64);   // 128 output cols -> 2 col-groups of 4 waves
    dim3 gemmGridZ(n / 16, 1);            // 32 output cols  -> 2 active waves, 2 guarded
    int scatterBlocks = (E + 7) / 8;      // 8 waves per 256-thread block

    // ---- layer 1: h0(256) @ W1 -> t(128); aggregate -> g; relu
    wmma_gemm_f32<<<gemmGridH, 128, 0, stream>>>(h0, W1, t, DIM_IN, DIM_H);
    selfloop_bias_kernel<<<(int)((totH + 255) / 256), 256, 0, stream>>>(t, dis, b1, g, DIM_H, totH);
    edge_scatter_kernel<<<scatterBlocks, 256, 0, stream>>>(t, src, dst, dis, g, E, DIM_H);
    relu_kernel<<<(int)((totH + 255) / 256), 256, 0, stream>>>(g, totH);

    // ---- layer 2: g(128) @ W2 -> t(128); aggregate -> g; relu
    wmma_gemm_f32<<<gemmGridH, 128, 0, stream>>>(g, W2, t, DIM_H, DIM_H);
    selfloop_bias_kernel<<<(int)((totH + 255) / 256), 256, 0, stream>>>(t, dis, b2, g, DIM_H, totH);
    edge_scatter_kernel<<<scatterBlocks, 256, 0, stream>>>(t, src, dst, dis, g, E, DIM_H);
    relu_kernel<<<(int)((totH + 255) / 256), 256, 0, stream>>>(g, totH);

    // ---- layer 3: g(128) @ W3 -> t(32); aggregate -> d_out (no relu)
    wmma_gemm_f32<<<gemmGridZ, 128, 0, stream>>>(g, W3, t, DIM_H, DIM_Z);
    selfloop_bias_kernel<<<(int)((totZ + 255) / 256), 256, 0, stream>>>(t, dis, b3, out, DIM_Z, totZ);
    edge_scatter_kernel<<<scatterBlocks, 256, 0, stream>>>(t, src, dst, dis, out, E, DIM_Z);
}